// KernelConv_11261404250772
// MI455X (gfx1250) — compile-verified
//
#include <hip/hip_runtime.h>

// KPN per-pixel dynamic 5x5 conv: out = W * sum_kk core[kk] * shift(frames, kk)
// B=2, N=8, H=W=512, K=5. Memory-bound (AI ~0.45 flop/byte): optimize streaming.
// core (419 MB) streamed once with NT hints; frames tile staged via async
// global->LDS (ASYNCcnt path, b128 on interior tiles); wave32 blocks, float4 I/O.

#define KSZ  5
#define PADR 2
#define HDIM 512
#define WDIM 512
#define HW   (HDIM * WDIM)
#define NPLANES 16            // B*N planes
#define TW 64                 // tile width  (outputs)
#define TH 16                 // tile height (outputs)
#define LH (TH + 2 * PADR)    // 20 rows incl. halo
#define LW2 80                // staged cols: w0-4 .. w0+75 (16B aligned both ends)
#define LSTRIDE 84            // multiple of 4 floats -> 16B-aligned LDS rows for b128

typedef float f32x4 __attribute__((ext_vector_type(4)));
typedef int   i32x4v __attribute__((__vector_size__(16)));

#if defined(__has_builtin)
#if __has_builtin(__builtin_amdgcn_global_load_async_to_lds_b32)
#define HAS_ASYNC_LDS 1
#endif
#if __has_builtin(__builtin_amdgcn_global_load_async_to_lds_b128)
#define HAS_ASYNC_LDS128 1
#endif
#if __has_builtin(__builtin_amdgcn_s_wait_asynccnt)
#define HAS_WAIT_ASYNC 1
#endif
#endif

typedef __attribute__((address_space(1))) int*    as1_int_p;
typedef __attribute__((address_space(3))) int*    as3_int_p;
typedef __attribute__((address_space(1))) i32x4v* as1_v4i_p;
typedef __attribute__((address_space(3))) i32x4v* as3_v4i_p;

__device__ __forceinline__ void async_g2l_b32(const float* g, const float* l) {
#if defined(HAS_ASYNC_LDS)
  __builtin_amdgcn_global_load_async_to_lds_b32(
      __builtin_bit_cast(as1_int_p, (unsigned long long)g),
      __builtin_bit_cast(as3_int_p, (unsigned)(unsigned long long)l), 0, 0);
#else
  *(float*)l = *g;
#endif
}

__device__ __forceinline__ void async_g2l_b128(const float* g, const float* l) {
#if defined(HAS_ASYNC_LDS128)
  __builtin_amdgcn_global_load_async_to_lds_b128(
      __builtin_bit_cast(as1_v4i_p, (unsigned long long)g),
      __builtin_bit_cast(as3_v4i_p, (unsigned)(unsigned long long)l), 0, 0);
#else
  async_g2l_b32(g + 0, l + 0);
  async_g2l_b32(g + 1, l + 1);
  async_g2l_b32(g + 2, l + 2);
  async_g2l_b32(g + 3, l + 3);
#endif
}

__device__ __forceinline__ void wait_async_then_barrier() {
#if defined(HAS_ASYNC_LDS)
#if defined(HAS_WAIT_ASYNC)
  __builtin_amdgcn_s_wait_asynccnt(0);
#else
  asm volatile("s_wait_asynccnt 0" ::: "memory");
#endif
#endif
  __syncthreads();
}

__global__ __launch_bounds__(256) void kpn5x5_kernel(
    const float* __restrict__ frames,
    const float* __restrict__ core,
    const float* __restrict__ wgt,
    float* __restrict__ out) {
  __shared__ float smem[LH * LSTRIDE];

  const int p  = blockIdx.z;             // plane = b*N + n
  const int w0 = blockIdx.x * TW;
  const int h0 = blockIdx.y * TH;
  const int t  = threadIdx.x;

  const float* fplane = frames + (size_t)p * HW;

  // ---- Stage frames rows [h0-2, h0+17] x cols [w0-4, w0+75] into LDS. ----
  const bool interior = (h0 >= PADR) && (h0 + TH + PADR <= HDIM) &&
                        (w0 >= 4) && (w0 + (LW2 - 4) <= WDIM);
  if (interior) {
    // Fast path (70% of tiles): 400 async b128 copies, no bounds checks.
#pragma unroll
    for (int it = 0; it < 2; ++it) {
      const int idx = t + it * 256;                   // 0..399
      if (idx < LH * (LW2 / 4)) {
        const int r  = idx / (LW2 / 4);               // 0..19
        const int q4 = (idx - r * (LW2 / 4)) * 4;     // col in floats, mult of 4
        async_g2l_b128(fplane + (size_t)(h0 - PADR + r) * WDIM + (w0 - 4) + q4,
                       smem + r * LSTRIDE + q4);
      }
    }
  } else {
    // Boundary path: per-element async b32 with zero-fill outside the image.
    for (int idx = t; idx < LH * LW2; idx += 256) {
      const int r  = idx / LW2;
      const int c  = idx - r * LW2;
      const int gh = h0 - PADR + r;
      const int gw = w0 - 4 + c;
      if (((unsigned)gh < (unsigned)HDIM) && ((unsigned)gw < (unsigned)WDIM)) {
        async_g2l_b32(fplane + (size_t)gh * WDIM + gw, smem + r * LSTRIDE + c);
      } else {
        smem[r * LSTRIDE + c] = 0.0f;
      }
    }
  }
  wait_async_then_barrier();

  // ---- Each thread produces 4 consecutive pixels of one row (float4 I/O). ----
  const int tx = t & 15;                 // 16 threads span 64 px
  const int ty = t >> 4;                 // 16 rows
  const int h  = h0 + ty;
  const int wb = w0 + tx * 4;

  // core[b, n*25 + kk, h, w] -> plane-major stride HW; channel base = 25*p
  const float* cbase = core + (size_t)(25 * p) * HW + (size_t)h * WDIM + wb;

  f32x4 acc = {0.0f, 0.0f, 0.0f, 0.0f};
#pragma unroll
  for (int i = 0; i < KSZ; ++i) {
    // Per-i base pointer: j offsets are then <= 4*HW*4B = 4 MB, which fits the
    // signed 24-bit flat-load IOFFSET (+-8 MB) -> no per-load 64-bit adds.
    const float* crow = cbase + (size_t)(i * KSZ) * HW;

    // 8-float register window covers all 5 j-shifts for 4 pixels.
    // smem col c maps to image col (w0-4+c): need c = 4*tx + 2 + s.
    float f[8];
    const int sb = (ty + i) * LSTRIDE + tx * 4 + 2;
#pragma unroll
    for (int s = 0; s < 8; ++s) f[s] = smem[sb + s];
#pragma unroll
    for (int j = 0; j < KSZ; ++j) {
      // core is streamed exactly once (419 MB > 192 MB L2): non-temporal.
      f32x4 c4 = __builtin_nontemporal_load(
          (const f32x4*)(const void*)(crow + (size_t)j * HW));
      acc[0] += c4[0] * f[j + 0];
      acc[1] += c4[1] * f[j + 1];
      acc[2] += c4[2] * f[j + 2];
      acc[3] += c4[3] * f[j + 3];
    }
  }

  const size_t o = (size_t)p * HW + (size_t)h * WDIM + wb;
  f32x4 wv = *(const f32x4*)(const void*)(wgt + o);
  f32x4 r4 = wv * acc;
  __builtin_nontemporal_store(r4, (f32x4*)(void*)(out + o));
}

extern "C" void kernel_launch(void* const* d_in, const int* in_sizes, int n_in,
                              void* d_out, int out_size, void* d_ws, size_t ws_size,
                              hipStream_t stream) {
  const float* frames = (const float*)d_in[0];
  const float* core   = (const float*)d_in[1];
  const float* wgt    = (const float*)d_in[2];
  float* out          = (float*)d_out;

  dim3 grid(WDIM / TW, HDIM / TH, NPLANES);  // 8 x 32 x 16 = 4096 blocks
  dim3 block(256);                           // 8 wave32s per block
  hipLaunchKernelGGL(kpn5x5_kernel, grid, block, 0, stream, frames, core, wgt, out);
}